// UVTo3D_45062796870117
// MI455X (gfx1250) — compile-verified
//
#include <hip/hip_runtime.h>
#include <cmath>

// ---------------------------------------------------------------------------
// Barycentric spherical interpolation, MI455X (gfx1250, wave32).
//
//   Kernel 1: per-face precompute (vA, AB, AC, N=cross(AB,AC)) -> d_ws
//             (1280 faces x 12 floats = 60 KB; removes 6 sincos + 1 cross +
//              1 normalize per *point* vs the reference).
//   Kernel 2: stage the 60 KB face table into LDS via CDNA5 async
//             global_load_async_to_lds_b128 (ASYNCcnt + s_wait_asynccnt),
//             then a 2D-grid pass (blockIdx.y = batch -> no integer division)
//             over the 1.05M points: uv -> table lookup -> LDS face record ->
//             sincos point -> plane projection -> 3 crosses/norms ->
//             vertex gather -> weighted sum.
// ---------------------------------------------------------------------------

#define PI_F 3.14159265358979323846f
#define NFACE_MAX 1280
#define REC 12                      // floats per face record (48 B, 16B-aligned)

struct F3 { float x, y, z; };
__device__ __forceinline__ F3 cross3(F3 a, F3 b) {
    return { a.y*b.z - a.z*b.y, a.z*b.x - a.x*b.z, a.x*b.y - a.y*b.x };
}
__device__ __forceinline__ float dot3(F3 a, F3 b) { return a.x*b.x + a.y*b.y + a.z*b.z; }
__device__ __forceinline__ F3 sub3(F3 a, F3 b) { return { a.x-b.x, a.y-b.y, a.z-b.z }; }

__device__ __forceinline__ F3 uv_to_3d(float u, float v) {
    float phi = 2.0f * PI_F * (u - 0.5f);
    float th  = PI_F * (v - 0.5f);
    float sp, cp, st, ct;
    __sincosf(phi, &sp, &cp);
    __sincosf(th,  &st, &ct);
    return { ct * cp, ct * sp, st };
}

// ---- Kernel 1: per-face precompute --------------------------------------
__global__ void face_precompute(const int* __restrict__ faces,
                                const float* __restrict__ uv_verts,
                                float* __restrict__ ws, int F)
{
    int f = blockIdx.x * blockDim.x + threadIdx.x;
    if (f >= F) return;
    int ia = faces[f*3+0], ib = faces[f*3+1], ic = faces[f*3+2];
    F3 A = uv_to_3d(uv_verts[ia*2+0], uv_verts[ia*2+1]);
    F3 B = uv_to_3d(uv_verts[ib*2+0], uv_verts[ib*2+1]);
    F3 C = uv_to_3d(uv_verts[ic*2+0], uv_verts[ic*2+1]);
    F3 AB = sub3(B, A);
    F3 AC = sub3(C, A);
    F3 N  = cross3(AB, AC);         // |N| = areaBAC, N/|N| = plane normal
    float* r = ws + f * REC;
    r[0]=A.x;  r[1]=A.y;  r[2]=A.z;
    r[3]=AB.x; r[4]=AB.y; r[5]=AB.z;
    r[6]=AC.x; r[7]=AC.y; r[8]=AC.z;
    r[9]=N.x;  r[10]=N.y; r[11]=N.z;
}

// ---- Kernel 2: main interpolation (blockIdx.y = batch) -------------------
__global__ __launch_bounds__(256) void bary_interp(
    const float* __restrict__ verts,      // (B, V, 3)
    const float* __restrict__ uv,         // (B, HW, 2)
    const int*   __restrict__ faces,      // (F, 3)
    const int*   __restrict__ face_inds,  // (H, W)
    const float* __restrict__ ws,         // (F, REC) precomputed
    float*       __restrict__ out,        // (B, HW, 3)
    int HW, int V, int F, int H, int W)
{
    __shared__ float sFace[NFACE_MAX * REC];   // 60 KB face table in LDS

    // --- CDNA5 async copy: global face table -> LDS (ASYNCcnt path) ---
    {
        int chunks = F * (REC / 4);            // 16B chunks (REC=12 -> 3/face)
        const char* gbase = (const char*)ws;
        for (int j = threadIdx.x; j < chunks; j += blockDim.x) {
            unsigned    lds_addr = (unsigned)(size_t)(void*)&sFace[j * 4];
            const char* gaddr    = gbase + (size_t)j * 16;
            asm volatile("global_load_async_to_lds_b128 %0, %1, off"
                         :: "v"(lds_addr), "v"(gaddr) : "memory");
        }
        asm volatile("s_wait_asynccnt 0" ::: "memory");
    }
    __syncthreads();

    const float Wm1 = (float)(W - 1);
    const float Hm1 = (float)(H - 1);
    const int   b   = blockIdx.y;              // batch index from the grid

    // Uniform per-batch base pointers (SGPRs, hoisted out of the loop)
    const float2* __restrict__ uv2 = (const float2*)uv + (size_t)b * HW;
    const float*  __restrict__ vb  = verts + (size_t)b * V * 3;
    float*        __restrict__ ob  = out   + (size_t)b * HW * 3;

    const int stride = blockDim.x * gridDim.x;

    for (int p = blockIdx.x * blockDim.x + threadIdx.x; p < HW; p += stride) {
        if (p + stride < HW)
            __builtin_prefetch(uv2 + (p + stride), 0, 1);   // global_prefetch_b8

        float2 t = uv2[p];
        float u = t.x, v = t.y;

        // jnp.round = round-half-even -> __float2int_rn
        int ix = __float2int_rn(u * Wm1);
        int iy = __float2int_rn(v * Hm1);
        ix = min(max(ix, 0), W - 1);
        iy = min(max(iy, 0), H - 1);
        int f = face_inds[iy * W + ix];

        // face record from LDS (three ds_load_b128)
        const float4* r4 = (const float4*)&sFace[f * REC];
        float4 q0 = r4[0], q1 = r4[1], q2 = r4[2];
        F3 vA { q0.x, q0.y, q0.z };
        F3 AB { q0.w, q1.x, q1.y };
        F3 AC { q1.z, q1.w, q2.x };
        F3 N  { q2.y, q2.z, q2.w };

        // point on sphere
        F3 P   = uv_to_3d(u, v);
        F3 AP0 = sub3(P, vA);

        // remove component perpendicular to triangle plane
        float nn   = dot3(N, N);
        float rlen = rsqrtf(fmaxf(nn, 1e-24f));   // 1/areaBAC, also normalizes N
        F3 d  { N.x * rlen, N.y * rlen, N.z * rlen };
        float pr = dot3(d, AP0);
        F3 AP { AP0.x - d.x * pr, AP0.y - d.y * pr, AP0.z - d.z * pr };
        F3 BP = sub3(AP, AB);
        F3 BC = sub3(AC, AB);

        // sub-triangle areas -> barycentric weights
        F3 c1 = cross3(AB, AP);
        F3 c2 = cross3(AC, AP);
        F3 c3 = cross3(BC, BP);
        float wW = sqrtf(dot3(c1, c1)) * rlen;    // weight of C
        float wV = sqrtf(dot3(c2, c2)) * rlen;    // weight of B
        float wU = sqrtf(dot3(c3, c3)) * rlen;    // weight of A
        float inv_s = 1.0f / fmaxf(wU + wV + wW, 1e-12f);
        wU *= inv_s; wV *= inv_s; wW *= inv_s;

        // gather the 3 batch vertices and blend
        int ia = faces[f*3+0], ib = faces[f*3+1], ic = faces[f*3+2];
        float ox = wU * vb[ia*3+0] + wV * vb[ib*3+0] + wW * vb[ic*3+0];
        float oy = wU * vb[ia*3+1] + wV * vb[ib*3+1] + wW * vb[ic*3+1];
        float oz = wU * vb[ia*3+2] + wV * vb[ib*3+2] + wW * vb[ic*3+2];

        ob[3*p+0] = ox;
        ob[3*p+1] = oy;
        ob[3*p+2] = oz;
    }
}

// ---------------------------------------------------------------------------
extern "C" void kernel_launch(void* const* d_in, const int* in_sizes, int n_in,
                              void* d_out, int out_size, void* d_ws, size_t ws_size,
                              hipStream_t stream)
{
    const float* verts     = (const float*)d_in[0];   // (B, V, 3)
    const float* uv        = (const float*)d_in[1];   // (B, HW, 2)
    const float* uv_verts  = (const float*)d_in[2];   // (V, 2)
    const int*   faces     = (const int*)d_in[3];     // (F, 3)
    const int*   face_inds = (const int*)d_in[4];     // (H, W)
    float*       out       = (float*)d_out;
    float*       ws        = (float*)d_ws;

    // Derive shapes from flat sizes (reference: B=16, HW=65536, V=642, F=1280, H=W=1001)
    int V  = in_sizes[2] / 2;
    int F  = in_sizes[3] / 3;
    int B  = in_sizes[0] / (3 * V);
    int HW = in_sizes[1] / (2 * B);
    int HWmap = in_sizes[4];
    int W = (int)(std::sqrt((double)HWmap) + 0.5);
    int H = HWmap / W;
    if (F > NFACE_MAX) F = NFACE_MAX;   // LDS table capacity (matches reference F=1280)

    face_precompute<<<(F + 255) / 256, 256, 0, stream>>>(faces, uv_verts, ws, F);

    // 2D grid: x covers HW (4 points/thread), y = batch -> no div in kernel
    dim3 grid(64, (unsigned)B, 1);
    bary_interp<<<grid, 256, 0, stream>>>(verts, uv, faces, face_inds, ws, out,
                                          HW, V, F, H, W);
}